// SimpleQuantumLayer_3642132267809
// MI455X (gfx1250) — compile-verified
//
#include <hip/hip_runtime.h>

// Raw clang vector type so __builtin_nontemporal_load/store accept it
// (HIP's float4 is a struct and would be rejected by the builtin).
typedef float f32x4 __attribute__((ext_vector_type(4)));

#define INV_2PI 0.15915494309189535f

// Main streaming kernel: each lane handles one float4 = 2 batch rows.
//   in : [x0, x1] pairs (row-major BATCH x 2)
//   out: [c0, c0*c1] pairs
// cos via v_cos_f32 (hardware computes cos(2*pi*arg)), so we fold the
// 1/(2*pi) scale into an FMA: arg = x*INV_2PI + w*INV_2PI.
__global__ __launch_bounds__(256) void squant_vec4_kernel(
    const f32x4* __restrict__ in4,
    f32x4* __restrict__ out4,
    const float* __restrict__ w,
    int n4)
{
    int i = blockIdx.x * blockDim.x + threadIdx.x;
    if (i >= n4) return;

    // Uniform address -> scalar loads (s_load), broadcast to the wave.
    float w0s = w[0] * INV_2PI;
    float w1s = w[1] * INV_2PI;

    // Streaming load, TH=NT: single-use data, keep it out of L2 residency.
    f32x4 v = __builtin_nontemporal_load(&in4[i]);

    float c0 = __builtin_amdgcn_cosf(__builtin_fmaf(v.x, INV_2PI, w0s));
    float c1 = __builtin_amdgcn_cosf(__builtin_fmaf(v.y, INV_2PI, w1s));
    float c2 = __builtin_amdgcn_cosf(__builtin_fmaf(v.z, INV_2PI, w0s));
    float c3 = __builtin_amdgcn_cosf(__builtin_fmaf(v.w, INV_2PI, w1s));

    f32x4 r;
    r.x = c0;
    r.y = c0 * c1;
    r.z = c2;
    r.w = c2 * c3;

    __builtin_nontemporal_store(r, &out4[i]);
}

// Scalar tail for row counts not divisible by 2 rows/lane (not hit for the
// reference shape, but keeps the kernel shape-agnostic and deterministic).
__global__ __launch_bounds__(64) void squant_tail_kernel(
    const float* __restrict__ in,
    float* __restrict__ out,
    const float* __restrict__ w,
    int row_start, int n_rows)
{
    int r = row_start + blockIdx.x * blockDim.x + threadIdx.x;
    if (r >= n_rows) return;
    float w0s = w[0] * INV_2PI;
    float w1s = w[1] * INV_2PI;
    float c0 = __builtin_amdgcn_cosf(__builtin_fmaf(in[2 * r + 0], INV_2PI, w0s));
    float c1 = __builtin_amdgcn_cosf(__builtin_fmaf(in[2 * r + 1], INV_2PI, w1s));
    out[2 * r + 0] = c0;
    out[2 * r + 1] = c0 * c1;
}

extern "C" void kernel_launch(void* const* d_in, const int* in_sizes, int n_in,
                              void* d_out, int out_size, void* d_ws, size_t ws_size,
                              hipStream_t stream)
{
    const float* x = (const float*)d_in[0];          // (BATCH, 2) float32, flat
    const float* w = (const float*)d_in[1];          // (2,) float32
    float* out = (float*)d_out;                      // (BATCH, 2) float32, flat

    const long long total_floats = (long long)in_sizes[0];   // BATCH * 2
    const long long n_rows = total_floats / 2;

    // Vectorized body: 4 floats (2 rows) per lane.
    const long long n4 = total_floats / 4;
    if (n4 > 0) {
        const int block = 256;
        const long long grid = (n4 + block - 1) / block;
        squant_vec4_kernel<<<dim3((unsigned)grid), dim3(block), 0, stream>>>(
            (const f32x4*)x, (f32x4*)out, w, (int)n4);
    }

    // Tail rows (none for the reference shape: 16,777,216 % 4 == 0).
    const long long rows_done = n4 * 2;
    if (rows_done < n_rows) {
        const long long tail = n_rows - rows_done;
        const int block = 64;
        const long long grid = (tail + block - 1) / block;
        squant_tail_kernel<<<dim3((unsigned)grid), dim3(block), 0, stream>>>(
            x, out, w, (int)rows_done, (int)n_rows);
    }
}